// Mueller_23742579212608
// MI455X (gfx1250) — compile-verified
//
#include <hip/hip_runtime.h>
#include <hip/hip_bf16.h>

#define EPS_F 1e-6f

typedef __attribute__((ext_vector_type(4))) float v4f;

// Fast 1-ulp hardware transcendentals (v_rcp_f32 / v_sqrt_f32 / v_rsq_f32)
__device__ __forceinline__ float frcp(float x)  { return __builtin_amdgcn_rcpf(x);  }
__device__ __forceinline__ float fsqrt(float x) { return __builtin_amdgcn_sqrtf(x); }
__device__ __forceinline__ float frsq(float x)  { return __builtin_amdgcn_rsqf(x);  }

__global__ __launch_bounds__(256) void mueller_fresnel_kernel(
    const float* __restrict__ ci_in,
    const float* __restrict__ eta_in,
    float* __restrict__ out,
    int n) {
  int i = blockIdx.x * blockDim.x + threadIdx.x;
  if (i >= n) return;

  // Read-once inputs: non-temporal loads (TH=NT) keep them out of L2.
  float ci  = __builtin_nontemporal_load(ci_in + i);
  float eta = __builtin_nontemporal_load(eta_in + i);

  float rcp_eta = frcp(eta);                 // v_rcp_f32
  bool outside  = (ci >= 0.0f);
  float eta_it  = outside ? eta : rcp_eta;
  float eta_ti  = outside ? rcp_eta : eta;

  float cts    = 1.0f - eta_ti * eta_ti * (1.0f - ci * ci);
  float ci_abs = fabsf(ci);

  // sqrtz: complex sqrt of real (cts + EPS), then mulsign by sign(cts) (0 -> +1).
  float t = cts + EPS_F;
  float rt = fsqrt(fabsf(t));                // v_sqrt_f32, shared by both branches
  float ctt_re, ctt_im;
  if (t >= 0.0f) { ctt_re = rt;   ctt_im = 0.0f; }
  else           { ctt_re = 0.0f; ctt_im = rt;   }
  float sgn = (cts >= 0.0f) ? 1.0f : -1.0f;
  ctt_re *= sgn;
  ctt_im *= sgn;

  // a_s = (ci_abs - eta_it*ctt) / (ci_abs + eta_it*ctt)   (complex)
  float bre = eta_it * ctt_re, bim = eta_it * ctt_im;
  float asr, asi;
  {
    float xr = ci_abs - bre, xi = -bim;   // numerator
    float ur = ci_abs + bre, ui =  bim;   // denominator
    float inv = frcp(ur * ur + ui * ui);  // v_rcp_f32
    asr = (xr * ur + xi * ui) * inv;
    asi = (xi * ur - xr * ui) * inv;
  }

  // a_p = (eta_it*ci_abs - ctt) / (eta_it*ci_abs + ctt)   (complex)
  float C = eta_it * ci_abs;
  float apr, api;
  {
    float xr = C - ctt_re, xi = -ctt_im;
    float ur = C + ctt_re, ui =  ctt_im;
    float inv = frcp(ur * ur + ui * ui);  // v_rcp_f32
    apr = (xr * ur + xi * ui) * inv;
    api = (xi * ur - xr * ui) * inv;
  }

  // bad = (eta == 1) | (eta == 0)  ->  a_s = a_p = 0
  if (eta == 1.0f || eta == 0.0f) {
    asr = asi = apr = api = 0.0f;
  }

  // r_s = |a_s*a_s|, r_p = |a_p*a_p|
  float s2r = asr * asr - asi * asi, s2i = 2.0f * asr * asi;
  float p2r = apr * apr - api * api, p2i = 2.0f * apr * api;
  float r_s = fsqrt(s2r * s2r + s2i * s2i);  // v_sqrt_f32
  float r_p = fsqrt(p2r * p2r + p2i * p2i);  // v_sqrt_f32

  // sincos_arg_diff(a_p, a_s)
  float mag_s2 = asr * asr + asi * asi;   // |a_s|^2
  float mag_p2 = apr * apr + api * api;   // |a_p|^2
  float prod   = mag_p2 * mag_s2;
  float norm   = (prod == 0.0f) ? 0.0f : frsq(prod + EPS_F);  // v_rsq_f32
  float cos_delta = (apr * asr + api * asi) * norm;  // Re(a_p * conj(a_s)) * norm
  float sin_delta = (api * asr - apr * asi) * norm;  // Im(a_p * conj(a_s)) * norm

  float a_ = 0.5f * (r_s + r_p);
  float b_ = 0.5f * (r_s - r_p);
  float c_ = fsqrt(r_s * r_p + EPS_F);    // v_sqrt_f32; >= 1e-3 -> c==0 where is dead

  float cc = c_ * cos_delta;
  float cs = c_ * sin_delta;

  // Row-major 4x4 output, 64 B per ray: four b128 non-temporal stores.
  v4f v0 = {a_,   b_,   0.0f, 0.0f};
  v4f v1 = {b_,   a_,   0.0f, 0.0f};
  v4f v2 = {0.0f, 0.0f, cc,   -cs };
  v4f v3 = {0.0f, 0.0f, cs,   cc  };

  v4f* o = (v4f*)out + (size_t)i * 4;
  __builtin_nontemporal_store(v0, o + 0);
  __builtin_nontemporal_store(v1, o + 1);
  __builtin_nontemporal_store(v2, o + 2);
  __builtin_nontemporal_store(v3, o + 3);
}

extern "C" void kernel_launch(void* const* d_in, const int* in_sizes, int n_in,
                              void* d_out, int out_size, void* d_ws, size_t ws_size,
                              hipStream_t stream) {
  const float* ci  = (const float*)d_in[0];
  const float* eta = (const float*)d_in[1];
  float* out = (float*)d_out;
  int n = in_sizes[0];

  const int block = 256;            // 8 wave32 waves per block on CDNA5
  const int grid  = (n + block - 1) / block;
  mueller_fresnel_kernel<<<grid, block, 0, stream>>>(ci, eta, out, n);
}